// Prototype_13211319403320
// MI455X (gfx1250) — compile-verified
//
#include <hip/hip_runtime.h>

// CDNA5 / gfx1250: wave32, WMMA f32 16x16x4, async global->LDS (ASYNCcnt).
typedef float v2f __attribute__((ext_vector_type(2)));
typedef float v8f __attribute__((ext_vector_type(8)));
typedef int   v4i __attribute__((ext_vector_type(4)));

#define N_TOTAL 131072
#define D_DIM   1792
#define M_DIM   2048
#define K_TOP   16

#define ROWS_PER_BLOCK 64   // embed rows per workgroup (2 row-tiles x 2 subtiles)
#define COLS_PER_CHUNK 64   // centroid cols per M-chunk
#define KBLK           64   // K depth staged in LDS per stage
#define NCHUNKS (M_DIM / COLS_PER_CHUNK)  // 32
#define NKBLK   (D_DIM / KBLK)            // 28
#define PAD     4           // 68-float row stride: 16B-aligned, bank-conflict-free
#define LROW    (KBLK + PAD)

// ---- CDNA5 async global->LDS copy (ASYNCcnt path), with sync fallback ----
// Probe-confirmed signature: (AS1 v4i* gaddr, AS3 v4i* ldsaddr, imm offset, imm cpol)
#if __has_builtin(__builtin_amdgcn_global_load_async_to_lds_b128)
#define HAS_ASYNC_LDS 1
#define ASYNC_B128(gp, lp)                                                   \
    __builtin_amdgcn_global_load_async_to_lds_b128(                          \
        (__attribute__((address_space(1))) v4i*)(gp),                        \
        (__attribute__((address_space(3))) v4i*)(lp), 0, 0)
#define WAIT_ASYNC_8()  asm volatile("s_wait_asynccnt 0x8" ::: "memory")
#define WAIT_ASYNC_0()  asm volatile("s_wait_asynccnt 0x0" ::: "memory")
#else
#define HAS_ASYNC_LDS 0
#define ASYNC_B128(gp, lp)                                                   \
    do { *(float4*)(lp) = *(const float4*)(gp); } while (0)
#define WAIT_ASYNC_8()  do {} while (0)
#define WAIT_ASYNC_0()  do {} while (0)
#endif

// ---- Kernel 1: c_sq[m] = sum_k centroids[m][k]^2, one wave32 per row ----
__global__ __launch_bounds__(256) void csq_kernel(const float* __restrict__ cent,
                                                  float* __restrict__ csq) {
    const int wave = threadIdx.x >> 5;
    const int lane = threadIdx.x & 31;
    const int row  = blockIdx.x * 8 + wave;
    const float* p = cent + (size_t)row * D_DIM;
    float s = 0.0f;
    for (int k = lane; k < D_DIM; k += 32) { float v = p[k]; s += v * v; }
    #pragma unroll
    for (int off = 16; off > 0; off >>= 1) s += __shfl_xor(s, off, 32);
    if (lane == 0) csq[row] = s;
}

// ---- Kernel 2: fused (c_sq - 2 * E @ C^T) + per-row top-16 ----
__global__ __launch_bounds__(256) void dist_topk_kernel(const float* __restrict__ E,
                                                        const float* __restrict__ C,
                                                        const float* __restrict__ csq_g,
                                                        int* __restrict__ out) {
    // Double-buffered operand tiles + score tile.
    __shared__ float sE[2][ROWS_PER_BLOCK][LROW];              // 2 x 64 x 68
    __shared__ float sC[2][COLS_PER_CHUNK][LROW];              // 2 x 64 x 68
    __shared__ float sS[ROWS_PER_BLOCK][COLS_PER_CHUNK + PAD]; // 64 x 68

    const int tid  = threadIdx.x;
    const int lane = tid & 31;
    const int wave = tid >> 5;
    const int l16  = lane & 15;
    const int half = lane >> 4;      // which 16-lane half of the wave
    const int wr   = wave >> 2;      // wave row-tile: 0..1 (32 rows each)
    const int wc   = wave & 3;       // wave col-tile: 0..3 (16 cols each)
    const int r0   = blockIdx.x * ROWS_PER_BLOCK;

    // Register-resident running top-16 (meaningful for tid < 64: waves 0,1).
    float bestv[K_TOP];
    int   besti[K_TOP];
    #pragma unroll
    for (int t = 0; t < K_TOP; ++t) { bestv[t] = 3.0e38f; besti[t] = 0; }
    float vmax = 3.0e38f;
    int   pmax = 0;

    // Cooperative loader coordinates: 256 threads, 4 x b128 per array per stage.
    const int srow = tid >> 2;          // 0..63
    const int skq  = (tid & 3) * 16;    // 0,16,32,48

    for (int chunk = 0; chunk < NCHUNKS; ++chunk) {
        const int c0 = chunk * COLS_PER_CHUNK;
        v8f acc0 = {0.f, 0.f, 0.f, 0.f, 0.f, 0.f, 0.f, 0.f};
        v8f acc1 = acc0;

        // Prologue: stage k-block 0 into buffer 0 (8 async b128 per thread).
        {
            const float* ge = E + (size_t)(r0 + srow) * D_DIM + skq;
            const float* gc = C + (size_t)(c0 + srow) * D_DIM + skq;
            #pragma unroll
            for (int q = 0; q < 4; ++q) {
                ASYNC_B128(ge + 4 * q, &sE[0][srow][skq + 4 * q]);
                ASYNC_B128(gc + 4 * q, &sC[0][srow][skq + 4 * q]);
            }
        }

        for (int kb = 0; kb < NKBLK; ++kb) {
            const int cur = kb & 1;
            if (kb + 1 < NKBLK) {
                // Stage next k-block into the other buffer, then wait so that
                // only the 8 just-issued copies may remain outstanding
                // (async loads complete in order) -> current buffer is ready.
                const int nxt = cur ^ 1;
                const int k1  = (kb + 1) * KBLK;
                const float* ge = E + (size_t)(r0 + srow) * D_DIM + k1 + skq;
                const float* gc = C + (size_t)(c0 + srow) * D_DIM + k1 + skq;
                #pragma unroll
                for (int q = 0; q < 4; ++q) {
                    ASYNC_B128(ge + 4 * q, &sE[nxt][srow][skq + 4 * q]);
                    ASYNC_B128(gc + 4 * q, &sC[nxt][srow][skq + 4 * q]);
                }
                WAIT_ASYNC_8();
            } else {
                WAIT_ASYNC_0();
            }
            __syncthreads();   // everyone's copies for `cur` are visible

            // 32 WMMAs: two independent 16x16 accumulators share each B fetch.
            // A layout (16x4 f32): lane l16 -> row M=l16; VGPR0/1 hold K={0,1}
            // (lanes 0-15) / K={2,3} (lanes 16-31) => float2 at k + 2*half.
            #pragma unroll
            for (int kk = 0; kk < KBLK; kk += 4) {
                v2f b  = *(const v2f*)&sC[cur][wc * 16 + l16][kk + 2 * half];
                v2f a0 = *(const v2f*)&sE[cur][wr * 32 + l16][kk + 2 * half];
                v2f a1 = *(const v2f*)&sE[cur][wr * 32 + 16 + l16][kk + 2 * half];
                acc0 = __builtin_amdgcn_wmma_f32_16x16x4_f32(
                    false, a0, false, b, (short)0, acc0, false, false);
                acc1 = __builtin_amdgcn_wmma_f32_16x16x4_f32(
                    false, a1, false, b, (short)0, acc1, false, false);
            }
            __syncthreads();   // all readers done before `cur` is restaged
        }

        // Scores: s = c_sq[col] - 2*dot.  C/D layout: acc[v] lives at
        // (row = subtile_base + v + 8*half, col = wc*16 + l16).
        const float cs = csq_g[c0 + wc * 16 + l16];
        #pragma unroll
        for (int v = 0; v < 8; ++v) {
            sS[wr * 32 + v + 8 * half][wc * 16 + l16]      = cs - 2.0f * acc0[v];
            sS[wr * 32 + 16 + v + 8 * half][wc * 16 + l16] = cs - 2.0f * acc1[v];
        }
        __syncthreads();

        // Row owners (waves 0,1 fully active) merge 64 candidates each.
        if (tid < ROWS_PER_BLOCK) {
            for (int j = 0; j < COLS_PER_CHUNK; ++j) {
                const float cand = sS[tid][j];
                if (cand < vmax) {   // strict: keeps lowest index on ties
                    const int ci = c0 + j;
                    #pragma unroll
                    for (int t = 0; t < K_TOP; ++t)
                        if (t == pmax) { bestv[t] = cand; besti[t] = ci; }
                    vmax = bestv[0]; pmax = 0;
                    #pragma unroll
                    for (int t = 1; t < K_TOP; ++t)
                        if (bestv[t] > vmax) { vmax = bestv[t]; pmax = t; }
                }
            }
        }
        __syncthreads();   // sS stable until merge done
    }

    // Emit ascending-distance order (matches top_k of negated distance).
    if (tid < ROWS_PER_BLOCK) {
        int* op = out + (size_t)(r0 + tid) * K_TOP;
        #pragma unroll
        for (int j = 0; j < K_TOP; ++j) {
            float mv = bestv[0]; int mi = 0;
            #pragma unroll
            for (int t = 1; t < K_TOP; ++t)
                if (bestv[t] < mv) { mv = bestv[t]; mi = t; }
            int outi = 0;
            #pragma unroll
            for (int t = 0; t < K_TOP; ++t)
                if (t == mi) { outi = besti[t]; bestv[t] = 3.0e38f; }
            op[j] = outi;
        }
    }
}

extern "C" void kernel_launch(void* const* d_in, const int* in_sizes, int n_in,
                              void* d_out, int out_size, void* d_ws, size_t ws_size,
                              hipStream_t stream) {
    const float* E = (const float*)d_in[0];   // embeds   [N, D] fp32
    const float* C = (const float*)d_in[1];   // centroids[M, D] fp32
    // d_in[2] is K (==16), baked in at compile time.
    float* csq = (float*)d_ws;                // M floats of scratch
    int*   out = (int*)d_out;                 // [N, 16] int32 indices

    csq_kernel<<<M_DIM / 8, 256, 0, stream>>>(C, csq);
    dist_topk_kernel<<<N_TOTAL / ROWS_PER_BLOCK, 256, 0, stream>>>(E, C, csq, out);
}